// leaky_current_RNN_6442450944097
// MI455X (gfx1250) — compile-verified
//
#include <hip/hip_runtime.h>

typedef float v2f __attribute__((ext_vector_type(2)));
typedef float v8f __attribute__((ext_vector_type(8)));

#define B_    128
#define T_    512
#define I_    4
#define H_    512
#define O_    3
#define NREG  256
#define BT    16     // batch rows per workgroup (one WMMA M-tile)
#define TPW   4      // 16-col N-tiles per wave (8 waves * 4 = 32 tiles = 512 cols)
#define LSTR  516    // LDS row stride in floats; 516 % 64 == 4 -> bank-conflict-free fragments

// ---- init: hidden_states[0] = h, reg = 0 ----
__global__ void rnn_init_kernel(const float* __restrict__ h0,
                                float* __restrict__ hs,
                                float* __restrict__ regp) {
  int i = blockIdx.x * blockDim.x + threadIdx.x;
  if (i < B_ * H_) hs[i] = h0[i];
  if (i == 0) *regp = 0.0f;
}

// ---- pack W_rec into per-lane WMMA B-fragment order ----
// wpack[((nt*128 + kk)*32 + lane)*2 + j] = W_rec[(kk*4 + 2*(lane>>4) + j)*H + nt*16 + (lane&15)]
__global__ void rnn_pack_kernel(const float* __restrict__ Wrec,
                                float* __restrict__ wpack) {
  int gid = blockIdx.x * blockDim.x + threadIdx.x;   // 0 .. 262143
  int j    = gid & 1;
  int lane = (gid >> 1) & 31;
  int kk   = (gid >> 6) & 127;
  int nt   = gid >> 13;
  int k = kk * 4 + 2 * (lane >> 4) + j;
  int n = nt * 16 + (lane & 15);
  wpack[gid] = Wrec[k * H_ + n];
}

// ---- main recurrent kernel: 8 WGs x 256 threads, each WG owns 16 batch rows ----
__launch_bounds__(256, 1)
__global__ void rnn_main_kernel(const float* __restrict__ x,
                                const float* __restrict__ h0,
                                const float* __restrict__ Win,
                                const float* __restrict__ Wout,
                                const float* __restrict__ brec,
                                const float* __restrict__ bin,
                                const float* __restrict__ wpack,
                                float* __restrict__ outs,
                                float* __restrict__ hs,
                                float* __restrict__ regp) {
  __shared__ float hbuf[BT * LSTR];   // h_prev / h_new (own-column updates)
  __shared__ float abuf[BT * LSTR];   // tanh(h_prev), WMMA A staging
  __shared__ float xbuf[BT * I_];
  __shared__ float outred[4 * 48];
  __shared__ float sreg;

  const int tid  = threadIdx.x;
  const int lane = tid & 31;
  const int wave = tid >> 5;
  const int nl   = lane & 15;   // fragment column / A-row
  const int hi   = lane >> 4;   // lane-half (K/M split)
  const int b0   = blockIdx.x * BT;

  // prologue: h -> LDS
  for (int idx = tid; idx < BT * H_; idx += 256) {
    int r = idx >> 9, c = idx & (H_ - 1);
    hbuf[r * LSTR + c] = h0[(b0 + r) * H_ + c];
  }
  if (tid == 0) sreg = 0.0f;

  // hoisted per-tile constants: W_in columns, bias sum, reg-column mask
  float wi[TPW][I_];
  float bl[TPW];
  float rm[TPW];                    // 1.0 if tile columns < NREG else 0.0 (branchless reg)
  #pragma unroll
  for (int i = 0; i < TPW; ++i) {
    int n = (wave * TPW + i) * 16 + nl;
    bl[i] = brec[n] + bin[n];
    rm[i] = ((wave * TPW + i) * 16 < NREG) ? 1.0f : 0.0f;
    #pragma unroll
    for (int q = 0; q < I_; ++q) wi[i][q] = Win[q * H_ + n];
  }

  v8f zp[TPW];                      // z_prev tiles, kept in VGPRs
  #pragma unroll
  for (int i = 0; i < TPW; ++i)
    #pragma unroll
    for (int v = 0; v < 8; ++v) zp[i][v] = 0.0f;

  float regacc = 0.0f;
  const v2f* wp2 = (const v2f*)wpack;

  __syncthreads();

  for (int t = 0; t < T_; ++t) {
    const float flagf = (t > 0) ? 1.0f : 0.0f;   // z_prev term skipped at t=0

    // phase A: tanh(h_prev) -> abuf ; stage x_t
    for (int idx = tid; idx < BT * H_; idx += 256) {
      int r = idx >> 9, c = idx & (H_ - 1);
      abuf[r * LSTR + c] = tanhf(hbuf[r * LSTR + c]);
    }
    if (tid < BT * I_) {
      int r = tid >> 2, q = tid & 3;
      xbuf[tid] = x[(b0 + r) * (T_ * I_) + t * I_ + q];
    }
    __syncthreads();

    // phase B: z = tanh(h) @ W_rec + (bias_rec + bias_in + x_t @ W_in)
    #pragma unroll
    for (int i = 0; i < TPW; ++i) {
      const int nt = wave * TPW + i;
      const int n0 = nt * 16;
      v8f acc;
      #pragma unroll
      for (int v = 0; v < 8; ++v) {
        int M = v + 8 * hi;
        acc[v] = bl[i]
               + xbuf[M * 4 + 0] * wi[i][0] + xbuf[M * 4 + 1] * wi[i][1]
               + xbuf[M * 4 + 2] * wi[i][2] + xbuf[M * 4 + 3] * wi[i][3];
      }
      const float* arow = &abuf[nl * LSTR + 2 * hi];
      const v2f*   wrow = wp2 + (nt * 128) * 32 + lane;
      #pragma unroll 8
      for (int kk = 0; kk < 128; ++kk) {
        v2f a = *(const v2f*)(arow + kk * 4);   // ds_load_b64, conflict-free
        v2f b = wrow[kk * 32];                  // coalesced global_load_b64
        acc = __builtin_amdgcn_wmma_f32_16x16x4_f32(false, a, false, b,
                                                    (short)0, acc, false, false);
      }
      // leaky update + branchless reg accumulation + hidden-state write
      const float rmask = rm[i];
      #pragma unroll
      for (int v = 0; v < 8; ++v) {
        int M = v + 8 * hi;
        float z  = acc[v];
        float hp = hbuf[M * LSTR + n0 + nl];
        float hn = 0.5f * (hp + z);             // alpha = 0.5
        hbuf[M * LSTR + n0 + nl] = hn;
        hs[(size_t)(t + 1) * (B_ * H_) + (size_t)(b0 + M) * H_ + n0 + nl] = hn;
        float dh = hn - hp;
        float dz = z - zp[i][v];
        regacc += rmask * (dh * dh + flagf * (dz * dz));
        zp[i][v] = z;
      }
    }
    __syncthreads();

    // phase C: out = h_new @ W_out (16x3), k split 4 ways across thread quarters
    {
      int quarter = tid >> 6, sub = tid & 63;
      if (sub < 48) {
        int r = sub / 3, o = sub - r * 3;
        int kbase = quarter * 128;
        float s = 0.0f;
        #pragma unroll 4
        for (int k = 0; k < 128; ++k)
          s += hbuf[r * LSTR + kbase + k] * Wout[(kbase + k) * O_ + o];
        outred[quarter * 48 + sub] = s;
      }
    }
    __syncthreads();
    if (tid < 48) {
      int r = tid / 3, o = tid - r * 3;
      float s = outred[tid] + outred[48 + tid] + outred[96 + tid] + outred[144 + tid];
      outs[(b0 + r) * (T_ * O_) + t * O_ + o] = s;
    }
    // no barrier needed: next outred write is 2 barriers away
  }

  // epilogue: reduce reg across lanes -> LDS -> global (pre-scaled)
  #pragma unroll
  for (int m = 16; m >= 1; m >>= 1) regacc += __shfl_xor(regacc, m, 32);
  if (lane == 0) atomicAdd(&sreg, regacc);
  __syncthreads();
  if (tid == 0)
    atomicAdd(regp, sreg * (1.0f / (512.0f * 128.0f * 256.0f)));  // /T/B/n_reg
}

extern "C" void kernel_launch(void* const* d_in, const int* in_sizes, int n_in,
                              void* d_out, int out_size, void* d_ws, size_t ws_size,
                              hipStream_t stream) {
  const float* x    = (const float*)d_in[0];
  const float* h0   = (const float*)d_in[1];
  const float* Wrec = (const float*)d_in[2];
  const float* Win  = (const float*)d_in[3];
  const float* Wout = (const float*)d_in[4];
  const float* brec = (const float*)d_in[5];
  const float* bin  = (const float*)d_in[6];

  float* outs = (float*)d_out;                       // (B,T,O)
  float* hsb  = outs + (size_t)B_ * T_ * O_;         // (T+1,B,H)
  float* regp = hsb + (size_t)(T_ + 1) * B_ * H_;    // scalar
  float* wpack = (float*)d_ws;                       // 1 MB packed W_rec

  rnn_init_kernel<<<(B_ * H_ + 255) / 256, 256, 0, stream>>>(h0, hsb, regp);
  rnn_pack_kernel<<<(32 * 128 * 32 * 2) / 256, 256, 0, stream>>>(Wrec, wpack);
  rnn_main_kernel<<<B_ / BT, 256, 0, stream>>>(x, h0, Win, Wout, brec, bin,
                                               wpack, outs, hsb, regp);
}